// lightgcn_4260607558029
// MI455X (gfx1250) — compile-verified
//
#include <hip/hip_runtime.h>
#include <hip/hip_bf16.h>

typedef float v2f __attribute__((ext_vector_type(2)));
typedef float v8f __attribute__((ext_vector_type(8)));

#define EMBED 64
#define NLAYERS 3

// ---------------- utility kernels ----------------

__global__ void zero_f(float* __restrict__ p, int n) {
    int i = blockIdx.x * blockDim.x + threadIdx.x;
    if (i < n) p[i] = 0.0f;
}

__global__ void deg_kernel(const int* __restrict__ dst, float* __restrict__ deg, int E) {
    int e = blockIdx.x * blockDim.x + threadIdx.x;
    if (e < E) atomicAdd(&deg[dst[e]], 1.0f);
}

__global__ void dis_kernel(float* __restrict__ deg, int N) {
    int i = blockIdx.x * blockDim.x + threadIdx.x;
    if (i < N) {
        float d = deg[i];
        deg[i] = (d > 0.0f) ? rsqrtf(fmaxf(d, 1e-12f)) : 0.0f;
    }
}

__global__ void norm_kernel(const int* __restrict__ src, const int* __restrict__ dst,
                            const float* __restrict__ dis, float* __restrict__ nrm, int E) {
    int e = blockIdx.x * blockDim.x + threadIdx.x;
    if (e < E) nrm[e] = dis[src[e]] * dis[dst[e]];
}

__global__ void init_kernel(const float* __restrict__ emb, float* __restrict__ x,
                            float* __restrict__ out, float alpha, int n) {
    int i = blockIdx.x * blockDim.x + threadIdx.x;
    if (i < n) {
        float v = emb[i];
        x[i] = v;
        out[i] = alpha * v;
    }
}

// one thread per (edge, float4-chunk): gather x[src]*norm, scatter-add to xn[dst]
__global__ void scatter_kernel(const int* __restrict__ src, const int* __restrict__ dst,
                               const float* __restrict__ nrm,
                               const float* __restrict__ x, float* __restrict__ xn, int E) {
    int idx = blockIdx.x * blockDim.x + threadIdx.x;
    int e = idx >> 4;          // EMBED/4 = 16 chunks per edge
    int c = idx & 15;
    if (e >= E) return;
    int s = src[e], d = dst[e];
    float w = nrm[e];
    float4 v = ((const float4*)x)[s * 16 + c];
    float* base = &xn[d * EMBED + c * 4];
    atomicAdd(base + 0, v.x * w);
    atomicAdd(base + 1, v.y * w);
    atomicAdd(base + 2, v.z * w);
    atomicAdd(base + 3, v.w * w);
}

__global__ void accum_kernel(const float* __restrict__ xn, float* __restrict__ out,
                             float alpha, int n) {
    int i = blockIdx.x * blockDim.x + threadIdx.x;
    if (i < n) out[i] += alpha * xn[i];
}

// ---------------- MLP head via V_WMMA_F32_16X16X4_F32 ----------------
// One wave (32 threads) per tile of 16 labels.
// H (LDS): [16 rows][128 K] padded to 132 floats/row -> conflict-free frag reads.
// D = H[16x128] @ W1[128x64]: 32 K-steps x 4 N-tiles of 16x16x4 f32 WMMA.

__global__ __launch_bounds__(32)
void mlp_wmma_kernel(const int* __restrict__ lsrc, const int* __restrict__ ldst,
                     const float* __restrict__ label, const float* __restrict__ outE,
                     const float* __restrict__ W1, const float* __restrict__ b1,
                     const float* __restrict__ W2, const float* __restrict__ b2,
                     float* __restrict__ pred, float* __restrict__ lossAcc, int L) {
    __shared__ float H[16 * 132];

    const int lane = threadIdx.x;          // wave32
    const int tile = blockIdx.x;
    const int base = tile * 16;            // 16 labels per wave; L % 16 == 0

    // stage gathered activations: row r = [out[src_r] (64) | out[dst_r] (64)]
    for (int i = lane; i < 16 * 32; i += 32) {
        int row = i >> 5;                  // label row within tile
        int c   = i & 31;                  // float4 slot 0..31 within the 128-f row
        int node = (c < 16) ? lsrc[base + row] : ldst[base + row];
        float4 v = ((const float4*)outE)[node * 16 + (c & 15)];
        *(float4*)&H[row * 132 + c * 4] = v;
    }
    __syncthreads();

    const int half = lane >> 4;            // 0: K pair {0,1}; 1: K pair {2,3}
    const int mr   = lane & 15;            // A: row M;  B: col N within tile

    v8f acc[4];
    #pragma unroll
    for (int t = 0; t < 4; ++t) acc[t] = (v8f){0.f,0.f,0.f,0.f,0.f,0.f,0.f,0.f};

    #pragma unroll 4
    for (int kk = 0; kk < 32; ++kk) {
        const int klo = kk * 4 + half * 2;
        v2f a;
        a.x = H[mr * 132 + klo];
        a.y = H[mr * 132 + klo + 1];
        #pragma unroll
        for (int t = 0; t < 4; ++t) {
            const int n = t * 16 + mr;
            v2f b;
            b.x = W1[klo * EMBED + n];
            b.y = W1[(klo + 1) * EMBED + n];
            acc[t] = __builtin_amdgcn_wmma_f32_16x16x4_f32(
                false, a, false, b, (short)0, acc[t], false, false);
        }
    }

    // epilogue: bias + relu + dot(W2); element (m,n): m = r + 8*half, n = 16t + mr
    float partial[8];
    #pragma unroll
    for (int r = 0; r < 8; ++r) partial[r] = 0.0f;
    #pragma unroll
    for (int t = 0; t < 4; ++t) {
        const int n = t * 16 + mr;
        const float bias = b1[n];
        const float w2 = W2[n];
        #pragma unroll
        for (int r = 0; r < 8; ++r) {
            float hv = acc[t][r] + bias;
            hv = fmaxf(hv, 0.0f);
            partial[r] += hv * w2;
        }
    }
    // butterfly-reduce across the 16 lanes sharing each m
    #pragma unroll
    for (int off = 8; off >= 1; off >>= 1) {
        #pragma unroll
        for (int r = 0; r < 8; ++r)
            partial[r] += __shfl_xor(partial[r], off, 16);
    }

    if (mr == 0) {                         // lane 0 -> m=0..7, lane 16 -> m=8..15
        const float bb = b2[0];
        const int m0 = 8 * half;
        float lsum = 0.0f;
        #pragma unroll
        for (int r = 0; r < 8; ++r) {
            const int idx = base + m0 + r;
            float p = partial[r] + bb;
            pred[idx] = p;
            float d = p - label[idx];
            lsum += d * d;
        }
        atomicAdd(lossAcc, lsum);
    }
}

__global__ void finalize_kernel(const float* __restrict__ lossAcc,
                                float* __restrict__ out_loss, int L) {
    if (threadIdx.x == 0 && blockIdx.x == 0)
        out_loss[0] = lossAcc[0] / (float)L;
}

// ---------------- launcher ----------------

extern "C" void kernel_launch(void* const* d_in, const int* in_sizes, int n_in,
                              void* d_out, int out_size, void* d_ws, size_t ws_size,
                              hipStream_t stream) {
    const int* edge_index       = (const int*)d_in[0];   // [2, E]
    const int* edge_label_index = (const int*)d_in[1];   // [2, L]
    const float* edge_label     = (const float*)d_in[2]; // [L]
    const float* embedding      = (const float*)d_in[3]; // [N, 64]
    const float* W1             = (const float*)d_in[4]; // [128, 64]
    const float* b1             = (const float*)d_in[5]; // [64]
    const float* W2             = (const float*)d_in[6]; // [64, 1]
    const float* b2             = (const float*)d_in[7]; // [1]

    const int E = in_sizes[0] / 2;
    const int L = in_sizes[1] / 2;
    const int N = in_sizes[3] / EMBED;
    const float alpha = 1.0f / (NLAYERS + 1);

    const int* src  = edge_index;
    const int* dst  = edge_index + E;
    const int* lsrc = edge_label_index;
    const int* ldst = edge_label_index + L;

    // workspace layout (floats)
    float* ws   = (float*)d_ws;
    float* deg  = ws;                 // N  (becomes deg^-1/2 in place)
    float* nrm  = deg  + N;           // E
    float* xA   = nrm  + E;           // N*64
    float* xB   = xA   + (size_t)N * EMBED;   // N*64
    float* outE = xB   + (size_t)N * EMBED;   // N*64
    float* acc  = outE + (size_t)N * EMBED;   // 1

    float* pred     = (float*)d_out;  // [L]
    float* out_loss = pred + L;       // [1]

    const int T = 256;
    const int nNE = N * EMBED;

    zero_f<<<(N + T - 1) / T, T, 0, stream>>>(deg, N);
    zero_f<<<1, 32, 0, stream>>>(acc, 1);

    deg_kernel<<<(E + T - 1) / T, T, 0, stream>>>(dst, deg, E);
    dis_kernel<<<(N + T - 1) / T, T, 0, stream>>>(deg, N);
    norm_kernel<<<(E + T - 1) / T, T, 0, stream>>>(src, dst, deg, nrm, E);

    init_kernel<<<(nNE + T - 1) / T, T, 0, stream>>>(embedding, xA, outE, alpha, nNE);

    float* xcur = xA;
    float* xnew = xB;
    for (int l = 0; l < NLAYERS; ++l) {
        zero_f<<<(nNE + T - 1) / T, T, 0, stream>>>(xnew, nNE);
        const int work = E * 16;
        scatter_kernel<<<(work + T - 1) / T, T, 0, stream>>>(src, dst, nrm, xcur, xnew, E);
        accum_kernel<<<(nNE + T - 1) / T, T, 0, stream>>>(xnew, outE, alpha, nNE);
        float* t2 = xcur; xcur = xnew; xnew = t2;
    }

    mlp_wmma_kernel<<<L / 16, 32, 0, stream>>>(lsrc, ldst, edge_label, outE,
                                               W1, b1, W2, b2, pred, acc, L);
    finalize_kernel<<<1, 32, 0, stream>>>(acc, out_loss, L);
}